// QueryEmbedding_66752381714500
// MI455X (gfx1250) — compile-verified
//
#include <hip/hip_runtime.h>
#include <hip/hip_bf16.h>
#include <math.h>

// ---------------------------------------------------------------------------
// QueryEmbedding fused pipeline for gfx1250 (MI455X), fp32 WMMA path.
//   Kernel 1: tokens (fourier GEMM + embed gathers + patch MLP) -> X [32768,768]
//   Kernel 2: fused output MLP  out = gelu(X@W1+b1)@W2 + b2, in-place on d_out
// ---------------------------------------------------------------------------

typedef float v2f __attribute__((ext_vector_type(2)));
typedef float v8f __attribute__((ext_vector_type(8)));

#define NQ_TOTAL   32768      // B*N = 4*8192
#define EMBED      768
#define OMHID      3072
#define NFRAMES    48
#define IMG_HW     256

__device__ __forceinline__ float gelu_exact(float x) {
    return 0.5f * x * (1.0f + erff(x * 0.70710678118654752440f));
}

__device__ __forceinline__ v8f wmma4(v2f a, v2f b, v8f c) {
    // D = A(16x4 f32) * B(4x16 f32) + C(16x16 f32)
    return __builtin_amdgcn_wmma_f32_16x16x4_f32(
        /*neg_a=*/false, a, /*neg_b=*/false, b,
        /*c_mod=*/(short)0, c, /*reuse_a=*/false, /*reuse_b=*/false);
}

// ===========================================================================
// Kernel 1: per-block 16 queries -> token sum X[q][0:768]
// ===========================================================================
#define QB 16

__global__ __launch_bounds__(256) void k1_tokens(
    const float* __restrict__ query,     // [4,8192,5]
    const float* __restrict__ images,    // [4,3,48,256,256]
    const float* __restrict__ uv_w,      // [40,768]
    const float* __restrict__ uv_b,      // [768]
    const float* __restrict__ t_src_emb, // [48,768]
    const float* __restrict__ t_tgt_emb, // [48,768]
    const float* __restrict__ t_cam_emb, // [48,768]
    const float* __restrict__ pm_w1,     // [27,108]
    const float* __restrict__ pm_b1,     // [108]
    const float* __restrict__ pm_w2,     // [108,768]
    const float* __restrict__ pm_b2,     // [768]
    float* __restrict__ X)               // [32768,768]
{
    __shared__ float feats[QB][40];
    __shared__ float patch[QB][27];
    __shared__ float hmid[QB][108];
    __shared__ float us[QB], vs[QB];
    __shared__ int   tsrc[QB], ttgt[QB], tcam[QB], xs[QB], ys[QB];

    const int tid = threadIdx.x;
    const int q0  = blockIdx.x * QB;

    if (tid < QB) {
        int qg = q0 + tid;
        float u = query[qg * 5 + 0]; u = fminf(fmaxf(u, 0.f), 1.f);
        float v = query[qg * 5 + 1]; v = fminf(fmaxf(v, 0.f), 1.f);
        us[tid] = u;  vs[tid] = v;
        tsrc[tid] = min(max((int)rintf(query[qg * 5 + 2]), 0), NFRAMES - 1);
        ttgt[tid] = min(max((int)rintf(query[qg * 5 + 3]), 0), NFRAMES - 1);
        tcam[tid] = min(max((int)rintf(query[qg * 5 + 4]), 0), NFRAMES - 1);
        xs[tid] = (int)rintf(u * (float)(IMG_HW - 1));
        ys[tid] = (int)rintf(v * (float)(IMG_HW - 1));
    }
    __syncthreads();

    // Fourier features: [sin(up) cos(up) sin(vp) cos(vp)], 10 freqs each
    for (int idx = tid; idx < QB * 40; idx += 256) {
        int q = idx / 40, k = idx % 40;
        int sel = k / 10;                       // 0:sin-u 1:cos-u 2:sin-v 3:cos-v
        float base = (sel < 2) ? us[q] : vs[q];
        float ang  = 6.28318530717958647692f * base * (float)(1 << (k % 10));
        feats[q][k] = (sel & 1) ? cosf(ang) : sinf(ang);
    }

    // 3x3x3 patch gather (layout: (B,C,T,H,W); flat index = c*9 + dy*3 + dx)
    for (int idx = tid; idx < QB * 27; idx += 256) {
        int q = idx / 27, r = idx % 27;
        int c = r / 9, dy = (r % 9) / 3, dx = r % 3;
        int qg = q0 + q;
        int b  = qg >> 13;                      // / 8192
        int yi = min(max(ys[q] + dy - 1, 0), IMG_HW - 1);
        int xi = min(max(xs[q] + dx - 1, 0), IMG_HW - 1);
        long off = ((((long)b * 3 + c) * NFRAMES + tsrc[q]) * IMG_HW + yi) * IMG_HW + xi;
        patch[q][r] = images[off];
    }
    __syncthreads();

    // patch MLP layer 1: 27 -> 108, exact gelu
    for (int idx = tid; idx < QB * 108; idx += 256) {
        int q = idx / 108, j = idx % 108;
        float s = pm_b1[j];
        #pragma unroll 9
        for (int k = 0; k < 27; ++k) s += patch[q][k] * pm_w1[k * 108 + j];
        hmid[q][j] = gelu_exact(s);
    }
    __syncthreads();

    // token sum: each thread owns (query = tid&15, 48 consecutive columns)
    const int q  = tid & 15;
    const int co = (tid >> 4) * 48;
    const int qg = q0 + q;
    const int es = tsrc[q] * EMBED, et = ttgt[q] * EMBED, ec = tcam[q] * EMBED;

    float acc[48];
    #pragma unroll
    for (int j = 0; j < 48; ++j) {
        int c = co + j;
        acc[j] = uv_b[c] + pm_b2[c] +
                 t_src_emb[es + c] + t_tgt_emb[et + c] + t_cam_emb[ec + c];
    }
    for (int k = 0; k < 40; ++k) {
        float f = feats[q][k];
        const float* wrow = uv_w + k * EMBED + co;
        #pragma unroll
        for (int j = 0; j < 48; ++j) acc[j] += f * wrow[j];
    }
    for (int k = 0; k < 108; ++k) {
        float f = hmid[q][k];
        const float* wrow = pm_w2 + k * EMBED + co;
        #pragma unroll
        for (int j = 0; j < 48; ++j) acc[j] += f * wrow[j];
    }
    float* orow = X + (long)qg * EMBED + co;
    #pragma unroll
    for (int j = 0; j < 48; ++j) orow[j] = acc[j];
}

// ===========================================================================
// Kernel 2: fused output MLP, fp32 WMMA 16x16x4.
//   Block: 256 threads (8 waves), M-tile = 32 rows, chunk = 256 hidden cols.
//   In-place on X (each block owns its 32 rows exclusively).
// ===========================================================================
#define MT      32
#define HCHUNK  256
#define HSTRIDE 258    // pad: byte stride 1032 (8B aligned, banks spread)

__global__ __launch_bounds__(256) void k2_mlp(
    const float* __restrict__ w1,   // [768,3072]
    const float* __restrict__ b1,   // [3072]
    const float* __restrict__ w2,   // [3072,768]
    const float* __restrict__ b2,   // [768]
    float* __restrict__ X)          // [32768,768] in/out
{
    __shared__ float Hs[MT * HSTRIDE];          // 33 KB

    const int tid  = threadIdx.x;
    const int lane = tid & 31;
    const int wv   = tid >> 5;                  // wave 0..7
    const int lhi  = lane >> 4;                 // 0/1: K-halves (A,B), M+8 (C/D)
    const int llo  = lane & 15;                 // M (A) / N (B,C,D)
    const int m0   = blockIdx.x * MT;

    v8f accO[12];                               // 2 row-tiles x 6 col-tiles of out
    {
        v8f z = {};
        #pragma unroll
        for (int i = 0; i < 12; ++i) accO[i] = z;
    }

    // A-fragment row bases for GEMM1 (row-tile 0: rows m0..m0+15)
    const float* Arow0 = X + (long)(m0 + llo) * EMBED + 2 * lhi;
    const float* Arow1 = Arow0 + (long)16 * EMBED;

    for (int chunk = 0; chunk < OMHID / HCHUNK; ++chunk) {
        const int n0 = chunk * HCHUNK;

        // ---- GEMM1: H[32][256] = X_tile[32][768] @ W1[:, n0:n0+256] ----
        v8f accH0, accH1, accH2, accH3;
        { v8f z = {}; accH0 = z; accH1 = z; accH2 = z; accH3 = z; }

        const int cw = n0 + wv * 32;            // this wave's 32 H-columns
        const float* Bbase = w1 + (long)(2 * lhi) * OMHID + cw + llo;

        for (int k = 0; k < EMBED; k += 4) {
            v2f a0 = *(const v2f*)(Arow0 + k);  // {X[m][k+2h], X[m][k+2h+1]}
            v2f a1 = *(const v2f*)(Arow1 + k);
            const float* bp = Bbase + (long)k * OMHID;
            v2f bv0, bv1;
            bv0.x = bp[0];            bv0.y = bp[OMHID];
            bv1.x = bp[16];           bv1.y = bp[OMHID + 16];
            accH0 = wmma4(a0, bv0, accH0);
            accH1 = wmma4(a0, bv1, accH1);
            accH2 = wmma4(a1, bv0, accH2);
            accH3 = wmma4(a1, bv1, accH3);
        }

        // bias + exact gelu -> LDS
        {
            float bia0 = b1[cw + llo];
            float bia1 = b1[cw + 16 + llo];
            int colL = wv * 32 + llo;           // chunk-local H column
            #pragma unroll
            for (int i = 0; i < 8; ++i) {
                int r0 = i + 8 * lhi;           // D layout: VGPR i -> row i + 8*hi
                Hs[r0 * HSTRIDE + colL]             = gelu_exact(accH0[i] + bia0);
                Hs[r0 * HSTRIDE + colL + 16]        = gelu_exact(accH1[i] + bia1);
                Hs[(16 + r0) * HSTRIDE + colL]      = gelu_exact(accH2[i] + bia0);
                Hs[(16 + r0) * HSTRIDE + colL + 16] = gelu_exact(accH3[i] + bia1);
            }
        }
        __syncthreads();

        // ---- GEMM2 accumulate: accO += gelu(H)[32][256] @ W2[n0:n0+256, ow:ow+96]
        const int ow = wv * 96;                 // this wave's 96 output columns
        const float* W2base = w2 + (long)(n0 + 2 * lhi) * EMBED + ow + llo;

        for (int k = 0; k < HCHUNK; k += 4) {
            v2f a0 = *(const v2f*)(&Hs[llo * HSTRIDE + k + 2 * lhi]);
            v2f a1 = *(const v2f*)(&Hs[(16 + llo) * HSTRIDE + k + 2 * lhi]);
            const float* w2r0 = W2base + (long)k * EMBED;
            const float* w2r1 = w2r0 + EMBED;
            #pragma unroll
            for (int ct = 0; ct < 6; ++ct) {
                v2f bb; bb.x = w2r0[ct * 16]; bb.y = w2r1[ct * 16];
                accO[ct]     = wmma4(a0, bb, accO[ct]);
                accO[6 + ct] = wmma4(a1, bb, accO[6 + ct]);
            }
        }
        __syncthreads();   // Hs reused next chunk
    }

    // ---- epilogue: + om_b2, store in place ----
    const int ow = wv * 96;
    #pragma unroll
    for (int ct = 0; ct < 6; ++ct) {
        int col  = ow + ct * 16 + llo;
        float bv = b2[col];
        #pragma unroll
        for (int i = 0; i < 8; ++i) {
            int r = i + 8 * lhi;
            X[(long)(m0 + r) * EMBED + col]      = accO[ct][i] + bv;
            X[(long)(m0 + 16 + r) * EMBED + col] = accO[6 + ct][i] + bv;
        }
    }
}

// ===========================================================================
extern "C" void kernel_launch(void* const* d_in, const int* in_sizes, int n_in,
                              void* d_out, int out_size, void* d_ws, size_t ws_size,
                              hipStream_t stream) {
    const float* query     = (const float*)d_in[0];
    const float* images    = (const float*)d_in[1];
    const float* uv_w      = (const float*)d_in[2];
    const float* uv_b      = (const float*)d_in[3];
    const float* t_src_emb = (const float*)d_in[4];
    const float* t_tgt_emb = (const float*)d_in[5];
    const float* t_cam_emb = (const float*)d_in[6];
    const float* pm_w1     = (const float*)d_in[7];
    const float* pm_b1     = (const float*)d_in[8];
    const float* pm_w2     = (const float*)d_in[9];
    const float* pm_b2     = (const float*)d_in[10];
    const float* om_w1     = (const float*)d_in[11];
    const float* om_b1     = (const float*)d_in[12];
    const float* om_w2     = (const float*)d_in[13];
    const float* om_b2     = (const float*)d_in[14];

    float* X = (float*)d_out;   // token sums, then overwritten with final output

    k1_tokens<<<NQ_TOTAL / QB, 256, 0, stream>>>(
        query, images, uv_w, uv_b, t_src_emb, t_tgt_emb, t_cam_emb,
        pm_w1, pm_b1, pm_w2, pm_b2, X);

    k2_mlp<<<NQ_TOTAL / MT, 256, 0, stream>>>(om_w1, om_b1, om_w2, om_b2, X);
}